// PNet_77017353552315
// MI455X (gfx1250) — compile-verified
//
#include <hip/hip_runtime.h>
#include <hip/hip_bf16.h>

typedef float v2f __attribute__((ext_vector_type(2)));
typedef float v8f __attribute__((ext_vector_type(8)));

#define NIN    16
#define NOUT   10
#define KG     64
#define BNW    64      // fourier width per feature
#define HH     8
#define DD     128
#define TILE   16      // batch rows per workgroup
#define NTHREADS 256   // 8 waves (wave32)
#define H2P    132     // padded LDS row stride for h2 (stride%64==4 -> conflict-free)
#define C1P    17      // padded row stride for per-wave C1 tile

__device__ __forceinline__ float sigf(float v) { return 1.0f / (1.0f + __expf(-v)); }

// D = A(16x4,f32) * B(4x16,f32) + C(16x16,f32), full-precision CDNA5 WMMA.
__device__ __forceinline__ v8f wmma4(v2f a, v2f b, v8f c) {
  return __builtin_amdgcn_wmma_f32_16x16x4_f32(false, a, false, b, (short)0, c,
                                               false, false);
}

// Intra-wave LDS ordering: DS ops are in-order per wave, so draining DScnt
// makes prior ds_stores visible to subsequent ds_loads without s_barrier.
__device__ __forceinline__ void lds_fence() {
  asm volatile("s_wait_dscnt 0" ::: "memory");
}

// butterfly reductions confined to each 16-lane half (masks < 16)
__device__ __forceinline__ float hsum16(float v) {
  v += __shfl_xor(v, 1, 32);
  v += __shfl_xor(v, 2, 32);
  v += __shfl_xor(v, 4, 32);
  v += __shfl_xor(v, 8, 32);
  return v;
}
__device__ __forceinline__ float hprod16(float v) {
  v *= __shfl_xor(v, 1, 32);
  v *= __shfl_xor(v, 2, 32);
  v *= __shfl_xor(v, 4, 32);
  v *= __shfl_xor(v, 8, 32);
  return v;
}

__global__ __launch_bounds__(NTHREADS)
void pnet_kernel(const float* __restrict__ x,   const float* __restrict__ w1,
                 const float* __restrict__ W2,  const float* __restrict__ b2,
                 const float* __restrict__ Wg1, const float* __restrict__ bg1,
                 const float* __restrict__ Wg2, const float* __restrict__ bg2,
                 const float* __restrict__ Wg3, const float* __restrict__ bg3,
                 const float* __restrict__ W4,  const float* __restrict__ b4,
                 float* __restrict__ outH4, float* __restrict__ outH3,
                 float* __restrict__ outH2, float* __restrict__ outGp)
{
  __shared__ float w2s[NIN * HH * BNW];   // 32 KB, L2-resident weights staged once
  __shared__ float w1s[BNW];
  __shared__ float b2s[NIN * HH];
  __shared__ float h2s[TILE * H2P];       // h2 tile [16 x 128] (padded)
  __shared__ float c1s[8][16 * C1P];      // per-wave GEMM1 output (padded)
  __shared__ float g2s[8][16 * 8];        // per-wave GEMM2 output, K padded 5->8
  __shared__ float h3s[TILE * KG];        // gated products

  const int tid      = threadIdx.x;
  const int lane     = tid & 31;
  const int wave     = tid >> 5;
  const int half     = lane >> 4;   // which 16-lane half
  const int ln       = lane & 15;
  const int lnc      = (ln < NOUT) ? ln : (NOUT - 1);  // clamped col (dead cols dup col 9)
  const int tileBase = blockIdx.x * TILE;

  // ---- phase 0: stage broadcast weights into LDS ----
  for (int i = tid; i < NIN * HH * BNW; i += NTHREADS) w2s[i] = W2[i];
  if (tid < BNW) w1s[tid] = w1[tid];
  if (tid < NIN * HH) b2s[tid] = b2[tid];
  __syncthreads();

  // ---- phase 1: fourier features + per-feature fc2 -> h2[16][128] ----
  {
    const int b = tid >> 4;       // batch row in tile
    const int n = tid & 15;       // feature
    const float xv = x[(size_t)(tileBase + b) * NIN + n];
    float acc[HH];
    #pragma unroll
    for (int h = 0; h < HH; ++h) acc[h] = b2s[n * HH + h];
    const float PI = 3.14159265358979323846f;
    #pragma unroll 4
    for (int d = 0; d < BNW; ++d) {
      const float ang = (float)(d & 31) * PI * xv * w1s[d];
      const float f = (d < 32) ? __cosf(ang) : __sinf(ang);
      #pragma unroll
      for (int h = 0; h < HH; ++h)
        acc[h] = fmaf(f, w2s[(n * HH + h) * BNW + d], acc[h]);
    }
    #pragma unroll
    for (int h = 0; h < HH; ++h) h2s[b * H2P + n * HH + h] = acc[h];
  }
  __syncthreads();

  // ---- phase 2: each wave handles 8 gates k ----
  float* c1w = &c1s[wave][0];
  float* g2w = &g2s[wave][0];
  const float* h2row = h2s + ln * H2P;          // A-frag row base (M = ln)

  for (int kk = 0; kk < 8; ++kk) {
    const int k = wave * 8 + kk;
    const float* Wg1k = Wg1 + (size_t)k * DD * NOUT;   // [128][10]
    const float* Wg3k = Wg3 + (size_t)k * 5 * DD;      // [5][128]

    // prefetch next gate's weights into cache (global_prefetch_b8), skipped on the wave's last gate
    if (kk < 7) {
      __builtin_prefetch(Wg1k + DD * NOUT + lane * 40, 0, 3);  // 5120B span
      __builtin_prefetch(Wg3k + 5 * DD + lane * 20, 0, 3);     // 2560B span
    }

    // GEMM1: C1[16x16] = h2[16x128] * Wg1k[128x16], 32 K-steps of 4.
    // B columns 10..15 duplicate column 9 (dead: GEMM2 reads only t<10).
    const float* b1base = Wg1k + 2 * half * NOUT + lnc;  // per-lane base, imm offsets below
    v8f c1 = {0.f, 0.f, 0.f, 0.f, 0.f, 0.f, 0.f, 0.f};
    #pragma unroll
    for (int s = 0; s < 32; ++s) {
      const int ac = 4 * s + 2 * half;       // A: M=ln, K split across halves
      v2f a, bb;
      a[0] = h2row[ac];
      a[1] = h2row[ac + 1];
      bb[0] = b1base[(4 * s) * NOUT];        // row 4s+2h, col lnc
      bb[1] = b1base[(4 * s) * NOUT + NOUT]; // row 4s+2h+1
      c1 = wmma4(a, bb, c1);
    }
    // bias (per column) + relu; columns >=10 hold junk but are never consumed
    {
      const float bgv = bg1[k * NOUT + lnc];
      #pragma unroll
      for (int r = 0; r < 8; ++r) c1[r] = fmaxf(c1[r] + bgv, 0.f);
    }
    // spill C1 to LDS (C layout: VGPR r -> row r / r+8, col = ln)
    #pragma unroll
    for (int r = 0; r < 8; ++r) c1w[(r + 8 * half) * C1P + ln] = c1[r];
    lds_fence();

    // GEMM2 (tiny 16x10x5): lanes 0..15, one output row each; K-pad cols 5..7 = 0
    if (half == 0) {
      #pragma unroll
      for (int s = 0; s < 5; ++s) {
        float a2 = bg2[k * 5 + s];
        #pragma unroll
        for (int t = 0; t < NOUT; ++t)
          a2 = fmaf(c1w[ln * C1P + t], Wg2[(size_t)(k * NOUT + t) * 5 + s], a2);
        g2w[ln * 8 + s] = fmaxf(a2, 0.f);
      }
      g2w[ln * 8 + 5] = 0.f;
      g2w[ln * 8 + 6] = 0.f;
      g2w[ln * 8 + 7] = 0.f;
    }
    lds_fence();

    // GEMM3: g3[16x128] = g2[16x8(Kpad)] * Wg3k[8(Kpad)x128], 2 K-steps, 8 N-tiles.
    // A K-columns 5..7 are exactly zero, so matching B rows may be garbage:
    //   s2=0 uses real rows {2h,2h+1}; s2=1 only needs row 4 (A col 4), so we
    //   splat row 4 into both fragment slots — fully branch-free.
    v8f acc[8];
    #pragma unroll
    for (int jn = 0; jn < 8; ++jn)
      acc[jn] = (v8f){0.f, 0.f, 0.f, 0.f, 0.f, 0.f, 0.f, 0.f};

    const float* b3a = Wg3k + 2 * half * DD + ln;  // rows 2h / 2h+1
    const float* b3b = Wg3k + 4 * DD + ln;         // row 4
    {
      v2f a0, a1;
      a0[0] = g2w[ln * 8 + 2 * half];
      a0[1] = g2w[ln * 8 + 2 * half + 1];
      a1[0] = g2w[ln * 8 + 4 + 2 * half];
      a1[1] = g2w[ln * 8 + 5 + 2 * half];
      #pragma unroll
      for (int jn = 0; jn < 8; ++jn) {
        v2f bb;
        bb[0] = b3a[16 * jn];
        bb[1] = b3a[DD + 16 * jn];
        acc[jn] = wmma4(a0, bb, acc[jn]);
      }
      #pragma unroll
      for (int jn = 0; jn < 8; ++jn) {
        const float r4 = b3b[16 * jn];
        v2f bb; bb[0] = r4; bb[1] = r4;     // slots beyond K=4 hit zero A columns
        acc[jn] = wmma4(a1, bb, acc[jn]);
      }
    }
    // bias bg3 (per column d = 16*jn + ln)
    #pragma unroll
    for (int jn = 0; jn < 8; ++jn) {
      const float bgv = bg3[(size_t)k * DD + 16 * jn + ln];
      #pragma unroll
      for (int r = 0; r < 8; ++r) acc[jn][r] += bgv;
    }

    // LayerNorm(128) per row, squash, write Gp, gated product -> h3
    #pragma unroll
    for (int r = 0; r < 8; ++r) {
      float sum = 0.f;
      #pragma unroll
      for (int jn = 0; jn < 8; ++jn) sum += acc[jn][r];
      const float mu = hsum16(sum) * (1.0f / 128.0f);
      float sq = 0.f;
      #pragma unroll
      for (int jn = 0; jn < 8; ++jn) {
        const float t = acc[jn][r] - mu;
        sq = fmaf(t, t, sq);
      }
      const float var = hsum16(sq) * (1.0f / 128.0f);
      const float rstd = rsqrtf(var + 1e-3f);
      #pragma unroll
      for (int jn = 0; jn < 8; ++jn) {
        const float gn = (acc[jn][r] - mu) * rstd;
        const float s6 = 6.0f * sigf(6.0f * gn);
        acc[jn][r] = sigf(10.0f * (s6 - 5.0f));
      }

      const int M = r + 8 * half;
      const size_t ob = (size_t)(tileBase + M) * (KG * DD) + (size_t)k * DD + ln;
      float p = 1.0f;
      #pragma unroll
      for (int jn = 0; jn < 8; ++jn) {
        const float g = acc[jn][r];
        outGp[ob + 16 * jn] = g;                               // 64B segments
        p *= fmaf(g, h2s[M * H2P + 16 * jn + ln], 1.0f);
      }
      p = hprod16(p);
      if (ln == 0) h3s[M * KG + k] = p;
    }
  }
  __syncthreads();

  // ---- phase 3: emit h2, h3, h4 ----
  for (int i = tid; i < TILE * DD; i += NTHREADS) {
    const int b = i / DD, c = i % DD;
    outH2[(size_t)(tileBase + b) * DD + c] = h2s[b * H2P + c];
  }
  for (int i = tid; i < TILE * KG; i += NTHREADS) {
    const int b = i / KG, c = i % KG;
    outH3[(size_t)(tileBase + b) * KG + c] = h3s[i];
  }
  if (tid < TILE * NOUT) {
    const int b = tid / NOUT, o = tid % NOUT;
    float a4 = b4[o];
    #pragma unroll 8
    for (int kx = 0; kx < KG; ++kx)
      a4 = fmaf(h3s[b * KG + kx], W4[kx * NOUT + o], a4);
    outH4[(size_t)(tileBase + b) * NOUT + o] = a4;
  }
}

extern "C" void kernel_launch(void* const* d_in, const int* in_sizes, int n_in,
                              void* d_out, int out_size, void* d_ws, size_t ws_size,
                              hipStream_t stream) {
  const float* x   = (const float*)d_in[0];
  const float* w1  = (const float*)d_in[1];
  const float* W2  = (const float*)d_in[2];
  const float* b2  = (const float*)d_in[3];
  const float* Wg1 = (const float*)d_in[4];
  const float* bg1 = (const float*)d_in[5];
  const float* Wg2 = (const float*)d_in[6];
  const float* bg2 = (const float*)d_in[7];
  const float* Wg3 = (const float*)d_in[8];
  const float* bg3 = (const float*)d_in[9];
  const float* W4  = (const float*)d_in[10];
  const float* b4  = (const float*)d_in[11];

  const int batch = in_sizes[0] / NIN;      // 16384
  float* out   = (float*)d_out;             // h4 | h3 | h2 | Gp, flat in order
  float* outH4 = out;
  float* outH3 = outH4 + (size_t)batch * NOUT;
  float* outH2 = outH3 + (size_t)batch * KG;
  float* outGp = outH2 + (size_t)batch * DD;

  const int nblocks = batch / TILE;         // 1024
  pnet_kernel<<<nblocks, NTHREADS, 0, stream>>>(
      x, w1, W2, b2, Wg1, bg1, Wg2, bg2, Wg3, bg3, W4, b4,
      outH4, outH3, outH2, outGp);
}